// CombinedLossSSIMCharbMSE_81372450390186
// MI455X (gfx1250) — compile-verified
//
#include <hip/hip_runtime.h>
#include <hip/hip_bf16.h>

typedef float v2f __attribute__((ext_vector_type(2)));
typedef float v8f __attribute__((ext_vector_type(8)));

// 11-tap Gaussian, sigma=1.5, normalized (matches reference _gaussian_kernel()).
__constant__ float c_w[11] = {
    0.00102838f, 0.00759876f, 0.03600077f, 0.10936070f, 0.21300582f,
    0.26601172f,
    0.21300582f, 0.10936070f, 0.03600077f, 0.00759876f, 0.00102838f};

#define WMMA4F32(A, B, C) \
    __builtin_amdgcn_wmma_f32_16x16x4_f32(false, (A), false, (B), (short)0, (C), false, false)

// Branchless zero-padded load. The halo is at most +/-5 rows/cols outside the
// 512x512 plane, so wrapping with &511 keeps the address inside the SAME
// plane's allocation (always safe to load); a cndmask zeroes OOB values.
// No EXEC manipulation -> loads can clause and overlap with WMMA issue.
__device__ __forceinline__ float ld_wrap(const float* __restrict__ p, int r, int cw,
                                         bool rowok, bool colok) {
    const float v = p[(r & 511) * 512 + cw];
    return (rowok & colok) ? v : 0.0f;
}

__global__ __launch_bounds__(128) void
CombinedLossSSIMCharbMSE_81372450390186_kernel(const float* __restrict__ pred,
                                               const float* __restrict__ target,
                                               float* __restrict__ out) {
    // per-wave V buffers: 5 fields x 16 rows x 32 cols (fp32), 4 waves
    __shared__ float lds[4 * 5 * 16 * 32];

    const int tid  = threadIdx.x;
    const int wave = tid >> 5;
    const int lane = tid & 31;
    const int half = lane >> 4;   // 0: lanes 0-15, 1: lanes 16-31
    const int m    = lane & 15;

    const int tile  = blockIdx.x * 4 + wave;     // 49152 tiles total
    const int plane = tile >> 10;                // 48 (n,c) planes
    const int tin   = tile & 1023;
    const int r0    = (tin >> 5) << 4;
    const int c0    = (tin & 31) << 4;

    const float* __restrict__ P = pred   + (size_t)plane * (512 * 512);
    const float* __restrict__ T = target + (size_t)plane * (512 * 512);
    float* __restrict__       O = out    + (size_t)plane * (512 * 512);

    // Band weights, identical per-lane values for the vertical-A and
    // horizontal-B band matrices:  band[s][q] = w[(4s + 2*half + q) - m]
    float band[7][2];
#pragma unroll
    for (int s = 0; s < 7; ++s) {
        const int k = 4 * s + 2 * half;
#pragma unroll
        for (int q = 0; q < 2; ++q) {
            const int d = k + q - m;
            band[s][q] = ((unsigned)d < 11u) ? c_w[d] : 0.0f;
        }
    }

    // ---------------- vertical pass: V = BandA(16x26) * X(26x16) ----------------
    v8f acc[2][5];
#pragma unroll
    for (int b = 0; b < 2; ++b)
#pragma unroll
        for (int f = 0; f < 5; ++f)
            acc[b][f] = (v8f){0.f, 0.f, 0.f, 0.f, 0.f, 0.f, 0.f, 0.f};

#pragma unroll
    for (int b = 0; b < 2; ++b) {
        const int  colraw = c0 - 5 + 16 * b + m;   // B-operand: N = m
        const bool colok  = (unsigned)colraw < 512u;
        const int  cw     = colraw & 511;
#pragma unroll
        for (int s = 0; s < 7; ++s) {
            const int  ra  = r0 - 5 + 4 * s + 2 * half;  // B-operand K rows
            const int  rb  = ra + 1;
            const bool oka = (unsigned)ra < 512u;
            const bool okb = (unsigned)rb < 512u;

            const float pa = ld_wrap(P, ra, cw, oka, colok);
            const float ta = ld_wrap(T, ra, cw, oka, colok);
            const float pb = ld_wrap(P, rb, cw, okb, colok);
            const float tb = ld_wrap(T, rb, cw, okb, colok);

            v2f a;
            a.x = band[s][0];
            a.y = band[s][1];
            const v2f Bp  = {pa, pb};
            const v2f Bt  = {ta, tb};
            const v2f Bpp = {pa * pa, pb * pb};
            const v2f Btt = {ta * ta, tb * tb};
            const v2f Bpt = {pa * ta, pb * tb};

            acc[b][0] = WMMA4F32(a, Bp,  acc[b][0]);
            acc[b][1] = WMMA4F32(a, Bt,  acc[b][1]);
            acc[b][2] = WMMA4F32(a, Bpp, acc[b][2]);
            acc[b][3] = WMMA4F32(a, Btt, acc[b][3]);
            acc[b][4] = WMMA4F32(a, Bpt, acc[b][4]);
        }
    }

    // spill V (C/D layout: row = j + 8*half, col = 16*b + m) to this wave's LDS slab
    float* __restrict__ myLds = lds + wave * (5 * 16 * 32);
#pragma unroll
    for (int f = 0; f < 5; ++f)
#pragma unroll
        for (int b = 0; b < 2; ++b)
#pragma unroll
            for (int j = 0; j < 8; ++j)
                myLds[f * 512 + (j + 8 * half) * 32 + 16 * b + m] = acc[b][f][j];

    asm volatile("s_wait_dscnt 0" ::: "memory");  // same-wave LDS RAW fence

    // ---------------- horizontal pass: OUT = V(16x26) * BandB(26x16) ----------------
    v8f hacc[5];
#pragma unroll
    for (int f = 0; f < 5; ++f)
        hacc[f] = (v8f){0.f, 0.f, 0.f, 0.f, 0.f, 0.f, 0.f, 0.f};

#pragma unroll
    for (int f = 0; f < 5; ++f) {
        const float* __restrict__ vrow = myLds + f * 512 + m * 32;  // A-operand: M = m
#pragma unroll
        for (int s = 0; s < 7; ++s) {
            const int k = 4 * s + 2 * half;
            v2f a;
            a.x = vrow[k];       // contiguous pair -> 2-address / b64 DS load
            a.y = vrow[k + 1];
            v2f bb;
            bb.x = band[s][0];
            bb.y = band[s][1];
            hacc[f] = WMMA4F32(a, bb, hacc[f]);
        }
    }

    // ---------------- elementwise SSIM / Charbonnier / MSE combine ----------------
    const float C1v = 0.0001f;   // (0.01)^2
    const float C2v = 0.0009f;   // (0.03)^2
#pragma unroll
    for (int j = 0; j < 8; ++j) {
        const int row = r0 + j + 8 * half;
        const int col = c0 + m;
        const int idx = row * 512 + col;
        const float p = P[idx];
        const float t = T[idx];

        const float mu_x = hacc[0][j];
        const float mu_y = hacc[1][j];
        const float gxx  = hacc[2][j];
        const float gyy  = hacc[3][j];
        const float gxy  = hacc[4][j];

        const float s2x = gxx - mu_x * mu_x;
        const float s2y = gyy - mu_y * mu_y;
        const float sxy = gxy - mu_x * mu_y;

        const float A1 = 2.0f * mu_x * mu_y + C1v;
        const float A2 = 2.0f * sxy + C2v;
        const float B1 = mu_x * mu_x + mu_y * mu_y + C1v;
        const float B2 = s2x + s2y + C2v;
        const float ssim = (A1 / B1) * (A2 / B2);

        const float diff  = p - t;
        const float charb = sqrtf(diff * diff + 1e-12f);   // EPS^2, EPS = 1e-6
        const float mse   = 20.0f * diff * diff;

        O[idx] = 0.3f * charb + 0.1f * mse + 0.6f * (1.0f - ssim);
    }
}

extern "C" void kernel_launch(void* const* d_in, const int* in_sizes, int n_in,
                              void* d_out, int out_size, void* d_ws, size_t ws_size,
                              hipStream_t stream) {
    (void)in_sizes; (void)n_in; (void)out_size; (void)d_ws; (void)ws_size;
    const float* pred   = (const float*)d_in[0];
    const float* target = (const float*)d_in[1];
    float*       out    = (float*)d_out;

    // 48 planes * 32x32 tiles = 49152 wave-tiles, 4 waves (128 threads) per block
    dim3 grid(12288), block(128);
    hipLaunchKernelGGL(CombinedLossSSIMCharbMSE_81372450390186_kernel,
                       grid, block, 0, stream, pred, target, out);
}